// Capsule_42502996361859
// MI455X (gfx1250) — compile-verified
//
#include <hip/hip_runtime.h>

// Capsule dynamic routing, fused, WMMA fp32 (gfx1250 / CDNA5, wave32).
//
// prior[d,b,n,o] = sum_i x[b,n,i] * W[d,n,i,o]    (never materialized)
// 3 routing iterations with softmax over n, squash over o.
//
// Block = (d, tile of 16 batches). 16 waves; each wave strides n.
// Per n: A = x[btile,n,0:8] (16x8), B = W[d,n,:,:] (8x16),
// two chained V_WMMA_F32_16X16X4_F32 give the 16x16 prior tile.
// s-pass folds the coupling coefficient into the A operand (row scaling),
// so the WMMA accumulator performs the sum over n directly.

#define DD 10
#define BB 256
#define NN 1152
#define II 8
#define OO 16
#define BTILE 16
#define NWAVES 16
#define NTHREADS (NWAVES * 32)

typedef float v2f __attribute__((ext_vector_type(2)));
typedef float v8f __attribute__((ext_vector_type(8)));

static __device__ __forceinline__ v8f wmma4(v2f a, v2f b, v8f c) {
  // D = A(16x4) * B(4x16) + C(16x16), fp32
  return __builtin_amdgcn_wmma_f32_16x16x4_f32(false, a, false, b,
                                               (short)0, c, false, false);
}

__global__ __launch_bounds__(NTHREADS) void caps_routing_kernel(
    const float* __restrict__ x,   // [B, N, I]
    const float* __restrict__ W,   // [D, N, I, O]
    float* __restrict__ out)       // [D, B, 1, 1, O]
{
  __shared__ float s_b[BTILE][NN + 1];   // routing logits (padded: kill bank conflicts)
  __shared__ float s_sacc[BTILE][OO];    // s accumulator (cross-wave)
  __shared__ float s_out[BTILE][OO];     // squashed output vector
  __shared__ float s_m[BTILE];           // softmax row max
  __shared__ float s_inv[BTILE];         // softmax row 1/sum

  const int tid  = threadIdx.x;
  const int lane = tid & 31;
  const int wave = tid >> 5;
  const int half = lane >> 4;        // lanes 0-15 vs 16-31
  const int mrow = lane & 15;        // A-matrix row (batch-in-tile); also D col (o)
  const int d    = blockIdx.x / (BB / BTILE);
  const int b0   = (blockIdx.x % (BB / BTILE)) * BTILE;

  // A operand layout (16x4 f32): lanes 0-15 hold K=0,1; lanes 16-31 hold K=2,3.
  const int kA = half * 2;
  const float* xrow  = x + (size_t)(b0 + mrow) * NN * II;   // + n*8 + k
  const float* wbase = W + (size_t)d * NN * (II * OO);      // + n*128 + k*16 + o

  // zero logits once; both b-passes then use +=
  for (int idx = tid; idx < BTILE * NN; idx += NTHREADS)
    s_b[idx / NN][idx % NN] = 0.0f;
  __syncthreads();

  // ---- s-pass: s[row,o] = sum_n c(row,n) * prior_n[row,o]; then squash ----
  auto pass_s = [&](int iter) {
    if (tid < BTILE * OO) s_sacc[tid >> 4][tid & 15] = 0.0f;
    const bool uniform = (iter == 0);
    const float rm   = uniform ? 0.0f : s_m[mrow];
    const float rinv = uniform ? 0.0f : s_inv[mrow];
    __syncthreads();
    v8f acc = {0.f, 0.f, 0.f, 0.f, 0.f, 0.f, 0.f, 0.f};
    for (int n = wave; n < NN; n += NWAVES) {
      const float cs = uniform ? (1.0f / (float)NN)
                               : __expf(s_b[mrow][n] - rm) * rinv;
      const float* xp = xrow + n * II;
      const float2 alo = *(const float2*)(xp + kA);
      const float2 ahi = *(const float2*)(xp + kA + 4);
      const float* wp = wbase + (size_t)n * (II * OO);
      v2f a0  = { alo.x * cs, alo.y * cs };
      v2f a1  = { ahi.x * cs, ahi.y * cs };
      v2f bb0 = { wp[(kA + 0) * OO + mrow], wp[(kA + 1) * OO + mrow] };
      v2f bb1 = { wp[(kA + 4) * OO + mrow], wp[(kA + 5) * OO + mrow] };
      acc = wmma4(a0, bb0, acc);   // K = 0..3
      acc = wmma4(a1, bb1, acc);   // K = 4..7, accumulates over n too
    }
    // C/D layout: VGPR r -> row r (lanes 0-15) / row r+8 (lanes 16-31), col = lane&15
#pragma unroll
    for (int r = 0; r < 8; ++r)
      atomicAdd(&s_sacc[r + half * 8][mrow], acc[r]);
    __syncthreads();
    if (wave == 0 && lane < BTILE) {
      float l2 = 0.0f;
#pragma unroll
      for (int o = 0; o < OO; ++o) { float v = s_sacc[lane][o]; l2 += v * v; }
      const float f = (l2 / (1.0f + l2)) * __frsqrt_rn(l2);
#pragma unroll
      for (int o = 0; o < OO; ++o) s_out[lane][o] = f * s_sacc[lane][o];
    }
    __syncthreads();
  };

  // ---- b-pass: b[row,n] += dot(prior_n[row,:], out[row,:]) ----
  auto pass_b = [&]() {
    float outv[8];
#pragma unroll
    for (int r = 0; r < 8; ++r) outv[r] = s_out[r + half * 8][mrow];
    for (int n = wave; n < NN; n += NWAVES) {
      const float* xp = xrow + n * II;
      const float2 alo = *(const float2*)(xp + kA);
      const float2 ahi = *(const float2*)(xp + kA + 4);
      const float* wp = wbase + (size_t)n * (II * OO);
      v2f a0  = { alo.x, alo.y };
      v2f a1  = { ahi.x, ahi.y };
      v2f bb0 = { wp[(kA + 0) * OO + mrow], wp[(kA + 1) * OO + mrow] };
      v2f bb1 = { wp[(kA + 4) * OO + mrow], wp[(kA + 5) * OO + mrow] };
      v8f p = {0.f, 0.f, 0.f, 0.f, 0.f, 0.f, 0.f, 0.f};
      p = wmma4(a0, bb0, p);
      p = wmma4(a1, bb1, p);
#pragma unroll
      for (int r = 0; r < 8; ++r) {
        float q = p[r] * outv[r];
        // reduce over the 16 lanes holding one row (xor<16 stays in-half)
        q += __shfl_xor(q, 1, 32);
        q += __shfl_xor(q, 2, 32);
        q += __shfl_xor(q, 4, 32);
        q += __shfl_xor(q, 8, 32);
        if (mrow == 0) s_b[r + half * 8][n] += q;
      }
    }
    __syncthreads();
  };

  // ---- softmax stats over n (one wave per batch row) ----
  auto pass_softmax = [&]() {
    const int row = wave;  // NWAVES == BTILE
    float m = -3.402823466e38f;
    for (int n = lane; n < NN; n += 32) m = fmaxf(m, s_b[row][n]);
#pragma unroll
    for (int s = 1; s < 32; s <<= 1) m = fmaxf(m, __shfl_xor(m, s, 32));
    float ssum = 0.0f;
    for (int n = lane; n < NN; n += 32) ssum += __expf(s_b[row][n] - m);
#pragma unroll
    for (int s = 1; s < 32; s <<= 1) ssum += __shfl_xor(ssum, s, 32);
    if (lane == 0) { s_m[row] = m; s_inv[row] = 1.0f / ssum; }
    __syncthreads();
  };

  // routing: 3 iterations, b updated after iters 0 and 1
  pass_s(0);        // uniform c = 1/N -> out0
  pass_b();         // b  = <prior, out0>
  pass_softmax();
  pass_s(1);        // -> out1
  pass_b();         // b += <prior, out1>
  pass_softmax();
  pass_s(2);        // -> out2 (final)

  if (tid < BTILE * OO) {
    const int row = tid >> 4, o = tid & 15;
    out[((size_t)d * BB + (b0 + row)) * OO + o] = s_out[row][o];
  }
}

extern "C" void kernel_launch(void* const* d_in, const int* in_sizes, int n_in,
                              void* d_out, int out_size, void* d_ws, size_t ws_size,
                              hipStream_t stream) {
  (void)in_sizes; (void)n_in; (void)d_ws; (void)ws_size; (void)out_size;
  const float* x = (const float*)d_in[0];   // [256,1152,8]
  const float* W = (const float*)d_in[1];   // [10,1152,8,16]
  float* out = (float*)d_out;               // [10,256,1,1,16]
  dim3 grid(DD * (BB / BTILE));             // 160 blocks
  dim3 block(NTHREADS);                     // 512 threads = 16 wave32
  caps_routing_kernel<<<grid, block, 0, stream>>>(x, W, out);
}